// StationFlowGCN2_63015760166989
// MI455X (gfx1250) — compile-verified
//
#include <hip/hip_runtime.h>
#include <hip/hip_bf16.h>

// ---------------------------------------------------------------------------
// StationFlowGCN2: 7-layer GCN on gfx1250 (MI455X).
//  - degree/rsqrt normalization computed ONCE (layer-invariant)
//  - dense GEMMs on the matrix pipe: V_WMMA_F32_16X16X4_F32 (exact fp32),
//    templated on (K, Nn) so all addresses fold to immediate offsets,
//    NT column-tiles per wave reuse each A fragment and interleave NT
//    independent WMMA chains (hides the WMMA->WMMA RAW hazard)
//  - edge aggregation: float4 gather + global fp32 atomic adds (L2-resident)
// ---------------------------------------------------------------------------

#define N_NODES 50000
#define N_EDGES 800000

typedef float v2f __attribute__((ext_vector_type(2)));
typedef float v8f __attribute__((ext_vector_type(8)));

__device__ __forceinline__ void atomAddF32(float* p, float v) {
  // agent-scope relaxed fadd -> global_atomic_add_f32 on gfx1250
  __hip_atomic_fetch_add(p, v, __ATOMIC_RELAXED, __HIP_MEMORY_SCOPE_AGENT);
}

// ---------------------------------------------------------------------------
// Degree / normalization precompute (runs once per launch)
// ---------------------------------------------------------------------------
__global__ void deg_init(float* __restrict__ deg) {
  int i = blockIdx.x * blockDim.x + threadIdx.x;
  if (i < N_NODES) deg[i] = 1.0f;  // self-loop weight
}

__global__ void deg_accum(const int* __restrict__ dst,
                          const float* __restrict__ w,
                          float* __restrict__ deg) {
  int e = blockIdx.x * blockDim.x + threadIdx.x;
  if (e < N_EDGES) atomAddF32(&deg[dst[e]], w[e]);
}

__global__ void deg_finalize(float* __restrict__ deg) {
  int i = blockIdx.x * blockDim.x + threadIdx.x;
  if (i < N_NODES) {
    float v = deg[i];
    deg[i] = (v > 0.0f) ? rsqrtf(v) : 0.0f;
  }
}

// ---------------------------------------------------------------------------
// GEMM: C[M,Nn] = A[M,K] @ B[K,Nn], fp32. One wave computes a 16 x (16*NT)
// strip of C. M = 50000 = 3125*16 exactly; K,Nn compile-time -> immediate
// address offsets, no tail code.
// A-tile layout (32-bit A 16x4): lanes 0-15 hold K=0,1 (v0,v1); lanes 16-31
// hold K=2,3. B-tile (4x16): one K-row striped across lanes per VGPR.
// C/D: 8 VGPRs; N = lane&15, M = v + 8*(lane>=16).
// ---------------------------------------------------------------------------
template <int K, int Nn, int NT>
__global__ __launch_bounds__(32) void gemm_wmma_f32(
    const float* __restrict__ A, const float* __restrict__ B,
    float* __restrict__ C) {
  const int lane = threadIdx.x & 31;
  const int half = lane >> 4;   // 0: K 0/1, 1: K 2/3 within each 4-slice
  const int r    = lane & 15;

  const float* Ap = A + (size_t)(blockIdx.x * 16 + r) * K + 2 * half;
  const float* Bp = B + blockIdx.y * (16 * NT) + r + 2 * half * Nn;

  v8f acc[NT];
#pragma unroll
  for (int t = 0; t < NT; ++t) acc[t] = (v8f){};

#pragma unroll 8
  for (int k = 0; k < K; k += 4) {
    const float2 av = *(const float2*)(Ap + k);   // K=k+2h, k+2h+1
    v2f a;
    a.x = av.x;
    a.y = av.y;
#pragma unroll
    for (int t = 0; t < NT; ++t) {
      v2f b;
      b.x = Bp[k * Nn + t * 16];          // row k+2h,   col tile t
      b.y = Bp[(k + 1) * Nn + t * 16];    // row k+2h+1, col tile t
      acc[t] = __builtin_amdgcn_wmma_f32_16x16x4_f32(
          false, a, false, b, (short)0, acc[t], false, false);
    }
  }

  float* Cp = C + (size_t)blockIdx.y * (16 * NT) + r;
#pragma unroll
  for (int t = 0; t < NT; ++t) {
#pragma unroll
    for (int v = 0; v < 8; ++v) {
      Cp[(size_t)(blockIdx.x * 16 + v + 8 * half) * Nn + t * 16] = acc[t][v];
    }
  }
}

// ---------------------------------------------------------------------------
// Aggregation: Acc[i,:] starts as self-loop term xw[i,:]*dinv[i]^2 (also the
// zero-init), then edges scatter xw[src]*norm into Acc[dst] via fp32 atomics.
// Each thread handles 4 channels (float4). cpeShift = log2(Dout/4).
// ---------------------------------------------------------------------------
__global__ void gcn_selfloop_init(const float* __restrict__ T,
                                  float* __restrict__ Acc,
                                  const float* __restrict__ dinv,
                                  int Dout, int cpeShift) {
  int tid = blockIdx.x * blockDim.x + threadIdx.x;
  int i = tid >> cpeShift;
  if (i >= N_NODES) return;
  int g = tid & ((1 << cpeShift) - 1);
  float s = dinv[i];
  s = s * s;
  float4 v = ((const float4*)(T + (size_t)i * Dout))[g];
  float4 o;
  o.x = v.x * s; o.y = v.y * s; o.z = v.z * s; o.w = v.w * s;
  ((float4*)(Acc + (size_t)i * Dout))[g] = o;
}

__global__ void gcn_scatter_edges(const float* __restrict__ T,
                                  float* __restrict__ Acc,
                                  const int* __restrict__ src,
                                  const int* __restrict__ dst,
                                  const float* __restrict__ w,
                                  const float* __restrict__ dinv,
                                  int Dout, int cpeShift) {
  int tid = blockIdx.x * blockDim.x + threadIdx.x;
  int e = tid >> cpeShift;
  if (e >= N_EDGES) return;
  int g = tid & ((1 << cpeShift) - 1);
  int s = src[e];
  int d = dst[e];
  float nrm = dinv[s] * w[e] * dinv[d];
  float4 v = ((const float4*)(T + (size_t)s * Dout))[g];
  float* out = Acc + (size_t)d * Dout + (g << 2);
  atomAddF32(out + 0, v.x * nrm);
  atomAddF32(out + 1, v.y * nrm);
  atomAddF32(out + 2, v.z * nrm);
  atomAddF32(out + 3, v.w * nrm);
}

__global__ void gcn_bias_relu(const float* __restrict__ Acc,
                              const float* __restrict__ bias,
                              float* __restrict__ Out,
                              int Dout, int cpeShift) {
  int tid = blockIdx.x * blockDim.x + threadIdx.x;
  int i = tid >> cpeShift;
  if (i >= N_NODES) return;
  int g = tid & ((1 << cpeShift) - 1);
  float4 v = ((const float4*)(Acc + (size_t)i * Dout))[g];
  float4 bb = ((const float4*)bias)[g];
  float4 o;
  o.x = fmaxf(v.x + bb.x, 0.0f);
  o.y = fmaxf(v.y + bb.y, 0.0f);
  o.z = fmaxf(v.z + bb.z, 0.0f);
  o.w = fmaxf(v.w + bb.w, 0.0f);
  ((float4*)(Out + (size_t)i * Dout))[g] = o;
}

// ---------------------------------------------------------------------------
extern "C" void kernel_launch(void* const* d_in, const int* in_sizes, int n_in,
                              void* d_out, int out_size, void* d_ws, size_t ws_size,
                              hipStream_t stream) {
  (void)in_sizes; (void)n_in; (void)out_size; (void)ws_size;

  const float* x  = (const float*)d_in[0];
  const int*   ei = (const int*)d_in[1];        // [2,E] row-major
  const float* ew = (const float*)d_in[2];
  const int* srcv = ei;
  const int* dstv = ei + N_EDGES;

  const float* W[7];
  const float* b[7];
  for (int i = 0; i < 7; ++i) {
    W[i] = (const float*)d_in[3 + 2 * i];
    b[i] = (const float*)d_in[4 + 2 * i];
  }
  const int dims[8] = {128, 256, 128, 64, 64, 32, 32, 32};

  // workspace layout: dinv[N] | bufH[N*256] | bufT[N*256]
  float* ws   = (float*)d_ws;
  float* dinv = ws;
  float* bufH = ws + ((N_NODES + 255) & ~255);
  float* bufT = bufH + (size_t)N_NODES * 256;

  // --- normalization, once ---
  deg_init<<<(N_NODES + 255) / 256, 256, 0, stream>>>(dinv);
  deg_accum<<<(N_EDGES + 255) / 256, 256, 0, stream>>>(dstv, ew, dinv);
  deg_finalize<<<(N_NODES + 255) / 256, 256, 0, stream>>>(dinv);

  const int MT = N_NODES / 16;  // 3125 row tiles, exact

  // --- 7 GCN layers ---
  for (int l = 0; l < 7; ++l) {
    const int Dout = dims[l + 1];
    const float* H = (l == 0) ? x : bufH;

    // GEMM on the matrix pipe: bufT = H @ W[l]  (shape-specialized)
    switch (l) {
      case 0: gemm_wmma_f32<128, 256, 4><<<dim3(MT, 256 / 64), 32, 0, stream>>>(H, W[l], bufT); break;
      case 1: gemm_wmma_f32<256, 128, 4><<<dim3(MT, 128 / 64), 32, 0, stream>>>(H, W[l], bufT); break;
      case 2: gemm_wmma_f32<128,  64, 4><<<dim3(MT, 1),        32, 0, stream>>>(H, W[l], bufT); break;
      case 3: gemm_wmma_f32< 64,  64, 4><<<dim3(MT, 1),        32, 0, stream>>>(H, W[l], bufT); break;
      case 4: gemm_wmma_f32< 64,  32, 2><<<dim3(MT, 1),        32, 0, stream>>>(H, W[l], bufT); break;
      default: gemm_wmma_f32<32,  32, 2><<<dim3(MT, 1),        32, 0, stream>>>(H, W[l], bufT); break;
    }

    const int cpe = Dout >> 2;            // float4 chunks per row
    int shift = 0;
    while ((1 << shift) < cpe) ++shift;   // log2(cpe), cpe is power of 2

    size_t tn = (size_t)N_NODES * cpe;
    gcn_selfloop_init<<<(unsigned)((tn + 255) / 256), 256, 0, stream>>>(
        bufT, bufH, dinv, Dout, shift);

    size_t te = (size_t)N_EDGES * cpe;
    gcn_scatter_edges<<<(unsigned)((te + 255) / 256), 256, 0, stream>>>(
        bufT, bufH, srcv, dstv, ew, dinv, Dout, shift);

    float* Out = (l == 6) ? (float*)d_out : bufH;
    gcn_bias_relu<<<(unsigned)((tn + 255) / 256), 256, 0, stream>>>(
        bufH, b[l], Out, Dout, shift);
  }
}